// GridPooling_6021544149810
// MI455X (gfx1250) — compile-verified
//
#include <hip/hip_runtime.h>
#include <hip/hip_bf16.h>
#include <float.h>

// ---------------- problem constants (from reference) ----------------
#define NPTS  524288
#define CIN   64
#define COUT  128
// stride=2, GRID=1024 -> pooled coords < 512 (9 bits each), batch < 4 (2 bits)
// compact key = gx | gy<<9 | gz<<18 | b<<27  (29 bits); numeric order of this
// key equals the sort order of the reference's 16-bit-field packed key.
#define KEY_BITS 29
#define NWORDS64 (1u << (KEY_BITS - 6))     // 8388608 u64 bitmap words
#define SCAN_CHUNK 8192                     // words per scan block
#define SCAN_BLOCKS (NWORDS64 / SCAN_CHUNK) // 1024

typedef __attribute__((ext_vector_type(2))) float v2f;
typedef __attribute__((ext_vector_type(8))) float v8f;
typedef __attribute__((ext_vector_type(4))) unsigned u32x4;
typedef __attribute__((ext_vector_type(8))) int i32x8;
typedef __attribute__((ext_vector_type(4))) int i32x4;
typedef __attribute__((address_space(3))) float lds_f;

// ---------------- d_out layout (float elements) ----------------
// [0, N*128)            seg_feat
// [N*128, N*131)        seg_coord
// [N*131, N*134)        grid_out (stored as float values)
// [N*134, N*135)        batch_out
// [N*135, N*136)        counts

// ---------------- small helpers ----------------
__device__ __forceinline__ void atomicMaxF(float* addr, float v) {
    // classic monotonic-bits trick; correct for finite floats and matches the
    // -FLT_MAX initialization of valid segments.
    if (v >= 0.0f) atomicMax((int*)addr, __float_as_int(v));
    else           atomicMin((unsigned int*)addr, __float_as_uint(v));
}

__device__ __forceinline__ void wait_asynccnt0() {
#if __has_builtin(__builtin_amdgcn_s_wait_asynccnt)
    __builtin_amdgcn_s_wait_asynccnt(0);
#else
    asm volatile("s_wait_asynccnt 0x0" ::: "memory");
#endif
}

// ---------------- kernels ----------------
__global__ void k_zero_bitmap(uint4* bm) {
    size_t i = (size_t)blockIdx.x * blockDim.x + threadIdx.x;
    bm[i] = make_uint4(0u, 0u, 0u, 0u);
}

__global__ void k_keys(const long long* __restrict__ gc,
                       const long long* __restrict__ batch,
                       const int* __restrict__ stridePtr,
                       unsigned* __restrict__ keys,
                       unsigned long long* __restrict__ bitmap) {
    int i = blockIdx.x * 256 + threadIdx.x;
    if (i >= NPTS) return;
    int stride = stridePtr[0];
    unsigned gx = (unsigned)(gc[(size_t)i * 3 + 0] / stride);
    unsigned gy = (unsigned)(gc[(size_t)i * 3 + 1] / stride);
    unsigned gz = (unsigned)(gc[(size_t)i * 3 + 2] / stride);
    unsigned b  = (unsigned)batch[i];
    unsigned key = gx | (gy << 9) | (gz << 18) | (b << 27);
    keys[i] = key;
    atomicOr(&bitmap[key >> 6], 1ull << (key & 63u));
}

__global__ void k_scan1(const unsigned long long* __restrict__ bitmap,
                        unsigned* __restrict__ blockSums) {
    __shared__ unsigned sh[256];
    unsigned t = threadIdx.x;
    size_t base = (size_t)blockIdx.x * SCAN_CHUNK + (size_t)t * 32;
    unsigned s = 0;
    for (int j = 0; j < 32; ++j) s += (unsigned)__popcll(bitmap[base + j]);
    sh[t] = s; __syncthreads();
    for (int off = 128; off > 0; off >>= 1) {
        if (t < (unsigned)off) sh[t] += sh[t + off];
        __syncthreads();
    }
    if (t == 0) blockSums[blockIdx.x] = sh[0];
}

__global__ void k_scan2(const unsigned* __restrict__ blockSums,
                        unsigned* __restrict__ blockOffs,
                        unsigned* __restrict__ numUniq) {
    __shared__ unsigned sh[SCAN_BLOCKS];
    unsigned t = threadIdx.x;
    unsigned v = blockSums[t];
    sh[t] = v; __syncthreads();
    for (int off = 1; off < SCAN_BLOCKS; off <<= 1) {
        unsigned x = (t >= (unsigned)off) ? sh[t - off] : 0u;
        __syncthreads();
        sh[t] += x;
        __syncthreads();
    }
    blockOffs[t] = sh[t] - v;                 // exclusive
    if (t == SCAN_BLOCKS - 1) numUniq[0] = sh[SCAN_BLOCKS - 1];
}

__global__ void k_scan3(const unsigned long long* __restrict__ bitmap,
                        const unsigned* __restrict__ blockOffs,
                        unsigned* __restrict__ wordPrefix,
                        unsigned* __restrict__ uniqKeys) {
    __shared__ unsigned sh[256];
    unsigned t = threadIdx.x;
    size_t base = (size_t)blockIdx.x * SCAN_CHUNK + (size_t)t * 32;
    unsigned s = 0;
    for (int j = 0; j < 32; ++j) s += (unsigned)__popcll(bitmap[base + j]);
    sh[t] = s; __syncthreads();
    for (int off = 1; off < 256; off <<= 1) {
        unsigned x = (t >= (unsigned)off) ? sh[t - off] : 0u;
        __syncthreads();
        sh[t] += x;
        __syncthreads();
    }
    unsigned run = blockOffs[blockIdx.x] + sh[t] - s;   // global exclusive prefix
    for (int j = 0; j < 32; ++j) {
        size_t w = base + j;
        unsigned long long bits = bitmap[w];
        wordPrefix[w] = run;
        while (bits) {
            int b = __builtin_ctzll(bits);
            uniqKeys[run++] = (unsigned)((w << 6) | (unsigned)b);
            bits &= bits - 1ull;
        }
    }
}

__global__ void k_init_out(float* __restrict__ out,
                           const unsigned* __restrict__ numUniq) {
    unsigned nu = numUniq[0];
    size_t i = (size_t)blockIdx.x * 256 + threadIdx.x;
    // seg_feat: -FLT_MAX for valid segments (segment_max identity), 0 otherwise
    unsigned seg = (unsigned)(i >> 7);
    out[i] = (seg < nu) ? -FLT_MAX : 0.0f;
    // zero the atomically-accumulated regions
    if (i < (size_t)NPTS * 3) out[(size_t)NPTS * COUT + i] = 0.0f;  // seg_coord
    if (i < (size_t)NPTS)     out[(size_t)NPTS * 135 + i] = 0.0f;   // counts
}

__global__ void k_cluster(const unsigned* __restrict__ keys,
                          const unsigned long long* __restrict__ bitmap,
                          const unsigned* __restrict__ wordPrefix,
                          const float* __restrict__ coord,
                          unsigned* __restrict__ clusterArr,
                          float* __restrict__ out) {
    int i = blockIdx.x * 256 + threadIdx.x;
    if (i >= NPTS) return;
    unsigned key = keys[i];
    size_t w = key >> 6;
    unsigned b = key & 63u;
    unsigned long long below = bitmap[w] & ((1ull << b) - 1ull);
    unsigned rank = wordPrefix[w] + (unsigned)__popcll(below);
    clusterArr[i] = rank;
    float* segc = out + (size_t)NPTS * COUT;
    atomicAdd(&segc[(size_t)rank * 3 + 0], coord[(size_t)i * 3 + 0]);
    atomicAdd(&segc[(size_t)rank * 3 + 1], coord[(size_t)i * 3 + 1]);
    atomicAdd(&segc[(size_t)rank * 3 + 2], coord[(size_t)i * 3 + 2]);
    atomicAdd(out + (size_t)NPTS * 135 + rank, 1.0f);
}

// Fused fp32 WMMA GEMM (feat @ W^T + bias) + segment-max via atomic float max.
// One wave computes a 16x128 output tile: 8 N-tiles x 16 K-steps of
// V_WMMA_F32_16X16X4_F32 (fp32 end-to-end, matching the reference precision).
// Weight (32 KB, contiguous 128x64 f32) is staged into LDS by the Tensor Data
// Mover (one DMA per block, TENSORcnt); bias is staged with async
// global->LDS loads (ASYNCcnt).
__global__ void __launch_bounds__(256)
k_gemm_segmax(const float* __restrict__ feat,
              const float* __restrict__ weight,
              const float* __restrict__ bias,
              const unsigned* __restrict__ clusterArr,
              float* __restrict__ seg_feat) {
    __shared__ float Wl[COUT * CIN];   // 32 KB of the 320 KB WGP LDS
    __shared__ float Bl[COUT];
    int tid = threadIdx.x;

#if __has_builtin(__builtin_amdgcn_tensor_load_to_lds)
    if (tid < 32) {
        // ---- Tensor DMA descriptor (D#) for the 128x64 f32 weight tile ----
        unsigned ldsA = (unsigned)(unsigned long long)(lds_f*)&Wl[0];
        unsigned long long ga = (unsigned long long)(const void*)weight;
        u32x4 g0;
        g0[0] = 1u;                                   // count=1 (valid user D#)
        g0[1] = ldsA;                                 // lds_addr
        g0[2] = (unsigned)(ga & 0xffffffffull);       // global_addr[31:0]
        g0[3] = (unsigned)((ga >> 32) & 0x1ffffffull) // global_addr[56:32]
                | (2u << 30);                         // type=2 ("image")
        i32x8 g1;
        g1[0] = (int)(2u << 16);       // data_size=4B; wg_mask=0 (not in cluster)
        g1[1] = (int)((unsigned)CIN << 16);   // tensor_dim0 = 64 (low 16 bits)
        g1[2] = (int)((unsigned)COUT << 16);  // dim0 hi=0 | tensor_dim1 = 128
        g1[3] = (int)((unsigned)CIN << 16);   // dim1 hi=0 | tile_dim0 = 64
        g1[4] = (int)COUT;                    // tile_dim1 = 128, tile_dim2 = 0
        g1[5] = (int)CIN;                     // tensor_dim0_stride = 64
        g1[6] = 0;
        g1[7] = 0;
        i32x4 z4 = {0, 0, 0, 0};
        i32x8 z8 = {0, 0, 0, 0, 0, 0, 0, 0};
        // 6-arg (clang-23 / therock) form: g0, g1, g2, g3, extra group, cpol
        __builtin_amdgcn_tensor_load_to_lds(g0, g1, z4, z4, z8, 0);
    }
    __builtin_amdgcn_s_wait_tensorcnt(0);
#else
    // fallback: async global->LDS copies (ASYNCcnt)
    for (int j = tid * 4; j < COUT * CIN; j += 256 * 4) {
        asm volatile("global_load_async_to_lds_b128 %0, %1, off"
                     :: "v"((lds_f*)&Wl[j]), "v"(weight + j) : "memory");
    }
#endif
    if (tid < 32) {
        // bias: 512 B = 32 lanes x b128 async copy to LDS
        asm volatile("global_load_async_to_lds_b128 %0, %1, off"
                     :: "v"((lds_f*)&Bl[tid * 4]), "v"(bias + (size_t)tid * 4)
                     : "memory");
    }
    wait_asynccnt0();
    __syncthreads();

    int wave = tid >> 5;
    int lane = tid & 31;
    int half = lane >> 4;      // K sub-block selector per WMMA layout
    int m    = lane & 15;      // row (A) / col (B,D) within tile
    int rowBase = (blockIdx.x * 8 + wave) * 16;

    // prefetch the next block's A rows into L2 (global_prefetch_b8)
    __builtin_prefetch(feat + (size_t)(rowBase + 8 * 16) * CIN, 0, 1);

    v8f acc[8] = {};
    const float* arow = feat + (size_t)(rowBase + m) * CIN;
    for (int k = 0; k < CIN; k += 4) {
        // A fragment (16x4 f32): lane half selects K pair {0,1} vs {2,3}
        v2f a;
        a.x = arow[k + 2 * half + 0];
        a.y = arow[k + 2 * half + 1];
#pragma unroll
        for (int nt = 0; nt < 8; ++nt) {
            // B fragment (4x16 f32) = W^T[k.., n..] = weight[n..][k..]
            const float* wl = &Wl[(nt * 16 + m) * CIN + k + 2 * half];
            v2f b;
            b.x = wl[0];
            b.y = wl[1];
            acc[nt] = __builtin_amdgcn_wmma_f32_16x16x4_f32(
                false, a, false, b, (short)0, acc[nt], false, false);
        }
    }

    // cluster ids for this tile's 16 rows: D layout row = base + j + half*8
    unsigned cl[8];
#pragma unroll
    for (int j = 0; j < 8; ++j) cl[j] = clusterArr[rowBase + half * 8 + j];

#pragma unroll
    for (int nt = 0; nt < 8; ++nt) {
        int col = nt * 16 + m;
        float bs = Bl[col];
#pragma unroll
        for (int j = 0; j < 8; ++j) {
            float v = acc[nt][j] + bs;
            atomicMaxF(&seg_feat[(size_t)cl[j] * COUT + col], v);
        }
    }
}

__global__ void k_final(const unsigned* __restrict__ uniqKeys,
                        const unsigned* __restrict__ numUniq,
                        float* __restrict__ out) {
    int s = blockIdx.x * 256 + threadIdx.x;
    if (s >= NPTS) return;
    unsigned nu = numUniq[0];
    float c = out[(size_t)NPTS * 135 + s];
    float inv = 1.0f / fmaxf(c, 1.0f);
    float* segc = out + (size_t)NPTS * COUT + (size_t)s * 3;
    segc[0] *= inv; segc[1] *= inv; segc[2] *= inv;
    float gx = 0.f, gy = 0.f, gz = 0.f, bb = 0.f;
    if ((unsigned)s < nu) {
        unsigned key = uniqKeys[s];
        gx = (float)(key & 511u);
        gy = (float)((key >> 9) & 511u);
        gz = (float)((key >> 18) & 511u);
        bb = (float)(key >> 27);
    }
    float* gout = out + (size_t)NPTS * 131 + (size_t)s * 3;
    gout[0] = gx; gout[1] = gy; gout[2] = gz;
    out[(size_t)NPTS * 134 + s] = bb;
}

// ---------------- launcher ----------------
extern "C" void kernel_launch(void* const* d_in, const int* in_sizes, int n_in,
                              void* d_out, int out_size, void* d_ws, size_t ws_size,
                              hipStream_t stream) {
    const float*     feat   = (const float*)d_in[0];
    const float*     coord  = (const float*)d_in[1];
    const float*     weight = (const float*)d_in[2];
    const float*     bias   = (const float*)d_in[3];
    const long long* gcoord = (const long long*)d_in[4];
    const long long* batch  = (const long long*)d_in[5];
    const int*       stride = (const int*)d_in[6];  // little-endian low word works for i32/i64
    float* out = (float*)d_out;

    // workspace carve-up
    char* ws = (char*)d_ws;
    unsigned long long* bitmap = (unsigned long long*)ws;            // 64 MB
    ws += (size_t)NWORDS64 * 8;
    unsigned* wordPrefix = (unsigned*)ws;                            // 32 MB
    ws += (size_t)NWORDS64 * 4;
    unsigned* blockSums = (unsigned*)ws;  ws += SCAN_BLOCKS * 4;
    unsigned* blockOffs = (unsigned*)ws;  ws += SCAN_BLOCKS * 4;
    unsigned* numUniq   = (unsigned*)ws;  ws += 16;
    unsigned* keys      = (unsigned*)ws;  ws += (size_t)NPTS * 4;
    unsigned* cluster   = (unsigned*)ws;  ws += (size_t)NPTS * 4;
    unsigned* uniqKeys  = (unsigned*)ws;  ws += (size_t)NPTS * 4;

    // 1) clear the 2^29-bit presence bitmap (uint4 stores)
    k_zero_bitmap<<<(NWORDS64 * 2) / 4 / 256, 256, 0, stream>>>((uint4*)bitmap);
    // 2) compact keys + bitmap set
    k_keys<<<NPTS / 256, 256, 0, stream>>>(gcoord, batch, stride, keys, bitmap);
    // 3) hierarchical popcount scan -> ranks + sorted-unique key list
    k_scan1<<<SCAN_BLOCKS, 256, 0, stream>>>(bitmap, blockSums);
    k_scan2<<<1, SCAN_BLOCKS, 0, stream>>>(blockSums, blockOffs, numUniq);
    k_scan3<<<SCAN_BLOCKS, 256, 0, stream>>>(bitmap, blockOffs, wordPrefix, uniqKeys);
    // 4) init outputs (-FLT_MAX for valid segments; zero accumulators)
    k_init_out<<<(NPTS * (size_t)COUT) / 256, 256, 0, stream>>>(out, numUniq);
    // 5) per-point cluster rank + coord/count accumulation
    k_cluster<<<NPTS / 256, 256, 0, stream>>>(keys, bitmap, wordPrefix, coord, cluster, out);
    // 6) fused fp32 WMMA projection + segment max (TDM/async-staged LDS)
    k_gemm_segmax<<<NPTS / (8 * 16), 256, 0, stream>>>(feat, weight, bias, cluster, out);
    // 7) mean divide + key decode
    k_final<<<NPTS / 256, 256, 0, stream>>>(uniqKeys, numUniq, out);
}